// RegionLoss_18975165513944
// MI455X (gfx1250) — compile-verified
//
#include <hip/hip_runtime.h>
#include <hip/hip_bf16.h>
#include <math.h>

// ---- problem constants (from reference) ----
#define NBATCH 32
#define NANCH  5
#define NCLS   80
#define NHH    76
#define NWW    76
#define MAXT   50
#define PLANE  (NHH*NWW)            // 5776
#define NCHAN  (NANCH*(5+NCLS))     // 425
#define GROUPS (PLANE/4)            // 1444 float4 groups per (b,a) plane
#define CELLS_PER_B (NANCH*GROUPS)  // 7220 threads-worth per batch
#define BLOCK  256
#define BLOCKS_PER_B ((CELLS_PER_B + BLOCK - 1)/BLOCK)  // 29
#define IOU_THRESH 0.6f
#define OBJ_SCALE_F 5.0f

__constant__ float c_anch[10] = {1.3221f, 1.73145f, 3.19275f, 4.00944f, 5.05587f,
                                 8.09892f, 9.47112f, 4.84053f, 11.2364f, 10.0071f};

// record layout (16 dwords / 64B):
//  f: 0 gxl, 1 gxr, 2 gyl, 3 gyr, 4 gw, 5 gh, 6 garea, 7 tx, 8 ty, 9 tw, 10 th
//  i: 11 gi, 12 gj, 13 bn, 14 valid, 15 cls
#define RECF 16

typedef __attribute__((ext_vector_type(4))) unsigned int uint4v;
typedef __attribute__((ext_vector_type(8))) int          int8v;
typedef __attribute__((ext_vector_type(4))) int          int4v;

__device__ __forceinline__ float iou_cc(float x1, float y1, float w1, float h1,
                                        float x2, float y2, float w2, float h2) {
    float mx = fminf(x1 - 0.5f*w1, x2 - 0.5f*w2);
    float Mx = fmaxf(x1 + 0.5f*w1, x2 + 0.5f*w2);
    float my = fminf(y1 - 0.5f*h1, y2 - 0.5f*h2);
    float My = fmaxf(y1 + 0.5f*h1, y2 + 0.5f*h2);
    float cw = w1 + w2 - (Mx - mx);
    float ch = h1 + h2 - (My - my);
    float inter = (cw <= 0.0f || ch <= 0.0f) ? 0.0f : cw * ch;
    float uni = w1*h1 + w2*h2 - inter;
    return inter / uni;
}

__device__ __forceinline__ float sigf(float v) { return 1.0f / (1.0f + expf(-v)); }

// ---------------- Kernel 1: per-GT preprocessing -> 64B corner-form records ----------------
__global__ void prep_kernel(const float* __restrict__ target, float* __restrict__ recs) {
    __shared__ int nz[MAXT];
    const int b = blockIdx.x;
    const int t = threadIdx.x;
    const float* tb = target + (size_t)b * MAXT * 5;
    float cls = 0.f, cx = 0.f, cy = 0.f, cw = 0.f, ch = 0.f;
    if (t < MAXT) {
        cls = tb[t*5+0]; cx = tb[t*5+1]; cy = tb[t*5+2]; cw = tb[t*5+3]; ch = tb[t*5+4];
        nz[t] = (cx != 0.0f) ? 1 : 0;
    }
    __syncthreads();
    if (t >= MAXT) return;
    int valid = 1;
    for (int s = 0; s <= t; ++s) valid &= nz[s];          // cumprod(cx != 0)

    float gx = cx * NWW, gy = cy * NHH, gw = cw * NWW, gh = ch * NHH;

    // best anchor by shape IoU (first max wins, like argmax)
    int bn = 0; float best = -1.0f;
    #pragma unroll
    for (int a = 0; a < NANCH; ++a) {
        float iou = iou_cc(0.f, 0.f, gw, gh, 0.f, 0.f, c_anch[2*a], c_anch[2*a+1]);
        if (iou > best) { best = iou; bn = a; }
    }
    int gi = (int)gx;  // trunc == floor for gx >= 0
    int gj = (int)gy;

    float* r = recs + ((size_t)b * MAXT + t) * RECF;
    int*   ri = (int*)r;
    r[0]  = gx - 0.5f * gw;   // gxl
    r[1]  = gx + 0.5f * gw;   // gxr
    r[2]  = gy - 0.5f * gh;   // gyl
    r[3]  = gy + 0.5f * gh;   // gyr
    r[4]  = gw;
    r[5]  = gh;
    r[6]  = gw * gh;          // garea
    r[7]  = gx - (float)gi;   // tx
    r[8]  = gy - (float)gj;   // ty
    r[9]  = logf(gw / c_anch[2*bn]);    // tw
    r[10] = logf(gh / c_anch[2*bn+1]);  // th
    ri[11] = gi; ri[12] = gj; ri[13] = bn; ri[14] = valid; ri[15] = (int)cls;
}

// ---------------- Kernel 2: fused per-cell loss ----------------
// grid = (BLOCKS_PER_B, NBATCH); each thread owns 4 consecutive cells of one (b,a) plane.
__global__ void __launch_bounds__(BLOCK)
loss_kernel(const float* __restrict__ out, const float* __restrict__ recs,
            float* __restrict__ partials) {
    __shared__ float srec[MAXT * RECF];   // 3200 B of GT records for this batch
    __shared__ float sred[BLOCK];
    const int b = blockIdx.y;

#if defined(__gfx1250__) && __has_builtin(__builtin_amdgcn_tensor_load_to_lds) && __has_builtin(__builtin_amdgcn_s_wait_tensorcnt)
    // CDNA5 Tensor Data Mover: async 1-D tile (800 dwords) global->LDS, issued by wave 0.
    if (threadIdx.x < 32) {
        unsigned lds_off = (unsigned)(unsigned long long)(void*)srec;          // LDS byte offset
        unsigned long long ga = (unsigned long long)(const void*)(recs + (size_t)b * MAXT * RECF);
        // D# group0: count=1 | lds_addr | global_addr[56:0] | type=2
        uint4v g0 = { 1u, lds_off, (unsigned)ga,
                      (unsigned)((ga >> 32) & 0x1ffffffull) | (2u << 30) };
        // D# group1: data_size=4B(code2); tensor_dim0=800; tile_dim0=800; stride0=800
        int8v g1 = { (int)(2u << 16),     // wg_mask=0, data_size=2
                     (int)(800u << 16),   // tensor_dim0[15:0] at bits 79:48
                     0,                   // tensor_dim0 hi / tensor_dim1 lo
                     (int)(800u << 16),   // tile_dim0 at bits 127:112
                     0,                   // tile_dim1=0, tile_dim2=0 (1-D)
                     800, 0, 0 };         // tensor_dim0_stride = 800
        int4v gz4 = { 0, 0, 0, 0 };
        int8v gz8 = { 0, 0, 0, 0, 0, 0, 0, 0 };
        __builtin_amdgcn_tensor_load_to_lds(g0, g1, gz4, gz4, gz8, 0);
        __builtin_amdgcn_s_wait_tensorcnt(0);   // TENSORcnt==0 before anyone reads LDS
    }
#else
    for (int k = threadIdx.x; k < MAXT * RECF; k += BLOCK)
        srec[k] = recs[(size_t)b * MAXT * RECF + k];
#endif
    __syncthreads();

    const int cid = blockIdx.x * BLOCK + threadIdx.x;
    float loss = 0.0f;
    if (cid < CELLS_PER_B) {
        const int a  = cid / GROUPS;
        const int g  = cid - a * GROUPS;
        const int p  = g * 4;             // plane-linear index of first cell
        const int j  = p / NWW;
        const int i0 = p - j * NWW;       // 76 % 4 == 0 -> same row for all 4 cells

        const size_t base = ((size_t)(b * NCHAN + a * 85)) * PLANE + p;
        const float4 v0 = *(const float4*)(out + base);
        const float4 v1 = *(const float4*)(out + base + (size_t)1 * PLANE);
        const float4 v2 = *(const float4*)(out + base + (size_t)2 * PLANE);
        const float4 v3 = *(const float4*)(out + base + (size_t)3 * PLANE);
        const float4 v4 = *(const float4*)(out + base + (size_t)4 * PLANE);
        const float o0[4] = {v0.x, v0.y, v0.z, v0.w};
        const float o1[4] = {v1.x, v1.y, v1.z, v1.w};
        const float o2[4] = {v2.x, v2.y, v2.z, v2.w};
        const float o3[4] = {v3.x, v3.y, v3.z, v3.w};
        const float o4[4] = {v4.x, v4.y, v4.z, v4.w};

        const float aw = c_anch[2*a], ah = c_anch[2*a+1];
        float xs[4], ys[4], wr[4], hr[4], cf[4];
        float pw[4], ph[4], pxl[4], pxr[4], pyl[4], pyr[4], pa[4];
        #pragma unroll
        for (int k = 0; k < 4; ++k) {
            xs[k] = sigf(o0[k]); ys[k] = sigf(o1[k]);
            wr[k] = o2[k];       hr[k] = o3[k];
            cf[k] = sigf(o4[k]);
            const float px = xs[k] + (float)(i0 + k);
            const float py = ys[k] + (float)j;
            pw[k]  = expf(wr[k]) * aw;
            ph[k]  = expf(hr[k]) * ah;
            pxl[k] = px - 0.5f * pw[k];  pxr[k] = px + 0.5f * pw[k];
            pyl[k] = py - 0.5f * ph[k];  pyr[k] = py + 0.5f * ph[k];
            pa[k]  = pw[k] * ph[k];
        }

        // max-IoU tracked as (inter, union) pair -> no division in the hot loop
        float bi[4] = {0.f, 0.f, 0.f, 0.f};
        float bu[4] = {1.f, 1.f, 1.f, 1.f};
        int   mt[4] = {-1, -1, -1, -1};
        for (int t = 0; t < MAXT; ++t) {
            const float* r  = srec + t * RECF;
            const int*   ri = (const int*)r;
            if (!ri[14]) continue;                         // invalid GT -> contributes 0
            const float gxl = r[0], gxr = r[1], gyl = r[2], gyr = r[3];
            const float gw = r[4], gh = r[5], garea = r[6];
            #pragma unroll
            for (int k = 0; k < 4; ++k) {
                const float mx = fminf(pxl[k], gxl), Mx = fmaxf(pxr[k], gxr);
                const float my = fminf(pyl[k], gyl), My = fmaxf(pyr[k], gyr);
                const float cw = (pw[k] + gw) - (Mx - mx);
                const float ch = (ph[k] + gh) - (My - my);
                const float inter = (cw <= 0.0f || ch <= 0.0f) ? 0.0f : cw * ch;
                const float uni = pa[k] + garea - inter;
                // inter/uni > bi/bu  <=>  inter*bu > bi*uni  (uni > 0 always)
                if (inter * bu[k] > bi[k] * uni) { bi[k] = inter; bu[k] = uni; }
            }
            if (ri[12] == j && ri[13] == a) {              // scatter match (last t wins)
                const int d = ri[11] - i0;
                if (d >= 0 && d < 4) mt[d] = t;
            }
        }

        #pragma unroll
        for (int k = 0; k < 4; ++k) {
            // max_iou > 0.6  <=>  bi > 0.6*bu
            float cmask = (bi[k] > IOU_THRESH * bu[k]) ? 0.0f : 1.0f;   // NOOBJ_SCALE = 1
            float tconf = 0.0f;
            if (mt[k] >= 0) {
                const float* r  = srec + mt[k] * RECF;
                const int*   ri = (const int*)r;
                cmask = OBJ_SCALE_F;
                const float dx = xs[k] - r[7], dy = ys[k] - r[8];
                const float dw = wr[k] - r[9], dh = hr[k] - r[10];
                loss += 0.5f * (dx*dx + dy*dy + dw*dw + dh*dh);   // COORD_SCALE = 1
                // exact tconf IoU (rare path, real division is fine here)
                {
                    const float mx = fminf(pxl[k], r[0]), Mx = fmaxf(pxr[k], r[1]);
                    const float my = fminf(pyl[k], r[2]), My = fmaxf(pyr[k], r[3]);
                    const float cw = (pw[k] + r[4]) - (Mx - mx);
                    const float ch = (ph[k] + r[5]) - (My - my);
                    const float inter = (cw <= 0.0f || ch <= 0.0f) ? 0.0f : cw * ch;
                    tconf = inter / (pa[k] + r[6] - inter);
                }
                // class CE via online logsumexp over 80 logits (only at matched cells)
                const int cls = ri[15];
                const size_t cb = base + (size_t)5 * PLANE + k;
                float m = -INFINITY, s = 0.0f, lc = 0.0f;
                for (int c = 0; c < NCLS; ++c) {
                    const float v = out[cb + (size_t)c * PLANE];
                    if (c == cls) lc = v;
                    if (v > m) { s = s * expf(m - v) + 1.0f; m = v; }
                    else       { s += expf(v - m); }
                }
                loss += -(lc - m - logf(s));                      // CLASS_SCALE = 1
            }
            const float dc = cf[k] - tconf;
            loss += 0.5f * cmask * dc * dc;                       // (conf*sq - tconf*sq)^2
        }
    }

    // deterministic block tree reduction
    sred[threadIdx.x] = loss;
    __syncthreads();
    for (int off = BLOCK / 2; off > 0; off >>= 1) {
        if (threadIdx.x < off) sred[threadIdx.x] += sred[threadIdx.x + off];
        __syncthreads();
    }
    if (threadIdx.x == 0) partials[b * BLOCKS_PER_B + blockIdx.x] = sred[0];
}

// ---------------- Kernel 3: deterministic final reduction ----------------
__global__ void reduce_kernel(const float* __restrict__ partials, int n, float* __restrict__ outv) {
    __shared__ double sd[256];
    double acc = 0.0;
    for (int k = threadIdx.x; k < n; k += 256) acc += (double)partials[k];
    sd[threadIdx.x] = acc;
    __syncthreads();
    for (int off = 128; off > 0; off >>= 1) {
        if (threadIdx.x < off) sd[threadIdx.x] += sd[threadIdx.x + off];
        __syncthreads();
    }
    if (threadIdx.x == 0) outv[0] = (float)sd[0];
}

extern "C" void kernel_launch(void* const* d_in, const int* in_sizes, int n_in,
                              void* d_out, int out_size, void* d_ws, size_t ws_size,
                              hipStream_t stream) {
    const float* output = (const float*)d_in[0];   // (32, 425, 76, 76) fp32
    const float* target = (const float*)d_in[1];   // (32, 250) fp32
    float* recs     = (float*)d_ws;                          // 32*50*16 floats = 100 KB
    float* partials = recs + (size_t)NBATCH * MAXT * RECF;   // 928 floats

    prep_kernel<<<NBATCH, 64, 0, stream>>>(target, recs);
    dim3 grid(BLOCKS_PER_B, NBATCH);
    loss_kernel<<<grid, BLOCK, 0, stream>>>(output, recs, partials);
    reduce_kernel<<<1, 256, 0, stream>>>(partials, NBATCH * BLOCKS_PER_B, (float*)d_out);
}